// Swing_38766374814073
// MI455X (gfx1250) — compile-verified
//
#include <hip/hip_runtime.h>

// Bandwidth-bound elementwise op (~520 MB total @ 23.3 TB/s ≈ 22 us floor).
// No contraction dimension -> WMMA inapplicable; the MI455X-relevant levers are
// b128 vector memory ops, *static* NT/RT temporal hints (stream is 235 MB >
// 192 MB L2, but ch9-11 are read twice -> keep them RT), and wave32 sizing.
// Channel classes are split into separate branch-free kernels so every load
// instruction carries exactly one (static) temporal hint.

typedef float v4f __attribute__((ext_vector_type(4)));

__device__ __forceinline__ v4f add001(v4f v) {
    v4f r;
#pragma unroll
    for (int i = 0; i < 4; ++i) r[i] = v[i] + 0.01f;   // lowers to v_pk_add_f32 x2
    return r;
}

// Channels {0..8, 13}: pure stream, no reuse anywhere -> NT load + NT store.
__global__ __launch_bounds__(256) void swing_passthru_nt(const float* __restrict__ x,
                                                         float* __restrict__ out,
                                                         unsigned int T) {
    const unsigned int cy = blockIdx.y;                  // 0..9
    const unsigned int ch = (cy < 9u) ? cy : 13u;        // skip 9,10,11,12
    const unsigned int t4 = (blockIdx.x * 256u + threadIdx.x) * 4u;
    if (t4 >= T) return;
    const size_t base = (size_t)ch * T + t4;
    const v4f v = __builtin_nontemporal_load((const v4f*)(x + base));
    __builtin_nontemporal_store(add001(v), (v4f*)(out + base));
}

// Channels 9,10,11: read again by the ch12 kernel -> RT load (cache in L2),
// output still streams out -> NT store.
__global__ __launch_bounds__(256) void swing_passthru_rt(const float* __restrict__ x,
                                                         float* __restrict__ out,
                                                         unsigned int T) {
    const unsigned int ch = 9u + blockIdx.y;             // 9..11
    const unsigned int t4 = (blockIdx.x * 256u + threadIdx.x) * 4u;
    if (t4 >= T) return;
    const size_t base = (size_t)ch * T + t4;
    const v4f v = *(const v4f*)(x + base);               // RT: reused by ch12 kernel
    __builtin_nontemporal_store(add001(v), (v4f*)(out + base));
}

// Channel 12: out12 = (upbeat(bartime) ? nudge(bpm,tsig) : x12) + 0.01
__global__ __launch_bounds__(256) void swing_ch12(const float* __restrict__ x,
                                                  float* __restrict__ out,
                                                  unsigned int T) {
    const unsigned int t4 = (blockIdx.x * 256u + threadIdx.x) * 4u;
    if (t4 >= T) return;
    const size_t base = (size_t)12u * T + t4;
    const v4f bpm = *(const v4f*)(x + (size_t)9u  * T + t4);   // RT: L2 hits
    const v4f tsg = *(const v4f*)(x + (size_t)10u * T + t4);
    const v4f bt  = *(const v4f*)(x + (size_t)11u * T + t4);
    const v4f x12 = __builtin_nontemporal_load((const v4f*)(x + base));
    v4f r;
#pragma unroll
    for (int i = 0; i < 4; ++i) {
        // jnp.mod(b, 0.5) == b - floor(b/0.5)*0.5 ; /0.5 == *2 exactly in FP
        const float m = bt[i] - __builtin_floorf(bt[i] * 2.0f) * 0.5f;
        const bool upbeat = __builtin_fabsf(m - 0.25f) < 0.05f;
        // (0.05f * 60000.0f) folds to the identical f32 constant JAX produces
        const float nudge = (0.05f * 60000.0f)
                            / __builtin_fmaxf(bpm[i], 0.001f) * tsg[i];
        r[i] = (upbeat ? nudge : x12[i]) + 0.01f;
    }
    __builtin_nontemporal_store(r, (v4f*)(out + base));
}

extern "C" void kernel_launch(void* const* d_in, const int* in_sizes, int n_in,
                              void* d_out, int out_size, void* d_ws, size_t ws_size,
                              hipStream_t stream) {
    (void)n_in; (void)out_size; (void)d_ws; (void)ws_size;
    const float* x = (const float*)d_in[0];
    float* out = (float*)d_out;
    const unsigned int T = (unsigned int)(in_sizes[0] / 14);   // 4194304
    const unsigned int nvec   = (T + 3u) / 4u;                 // float4s per channel
    const unsigned int blocks = (nvec + 255u) / 256u;
    const dim3 block(256, 1, 1);                               // 8 wave32 waves

    swing_passthru_nt<<<dim3(blocks, 10, 1), block, 0, stream>>>(x, out, T);
    swing_passthru_rt<<<dim3(blocks, 3,  1), block, 0, stream>>>(x, out, T);
    swing_ch12       <<<dim3(blocks, 1,  1), block, 0, stream>>>(x, out, T);
}